// NaiveLSTM_27839978013421
// MI455X (gfx1250) — compile-verified
//
#include <hip/hip_runtime.h>
#include <hip/hip_bf16.h>
#include <math.h>

// ---------------------------------------------------------------------------
// NaiveLSTM (degenerate: h/c never updated -> one big fused GEMM + epilogue)
//   inputs [32,512,1024] f32, 8x W[1024,1024], 8x b[1024]
//   out = concat(hidden_seq[512,32,1024], h_last[1,32,1024], c_last[1,32,1024])
// Main GEMM: native fp32 WMMA V_WMMA_F32_16X16X4_F32 (wave32), per-wave
// 32(M)x16(N) tile x 4 gates (8 accumulators) so each weight fragment feeds
// two WMMAs.  Epilogue uses hardware V_TANH_F32.
// ---------------------------------------------------------------------------

typedef __attribute__((ext_vector_type(2))) float v2f;
typedef __attribute__((ext_vector_type(8))) float v8f;

#define B_SZ   32
#define T_SZ   512
#define K_SZ   1024   // input size == hidden size
#define H_SZ   1024
#define HID_ELEMS ((size_t)T_SZ * B_SZ * H_SZ)  // 16777216

__device__ __forceinline__ float fast_tanhf(float x) {
#if __has_builtin(__builtin_amdgcn_tanhf)
    return __builtin_amdgcn_tanhf(x);          // v_tanh_f32
#elif __has_builtin(__builtin_amdgcn_tanh_f32)
    return __builtin_amdgcn_tanh_f32(x);
#else
    return tanhf(x);
#endif
}

__device__ __forceinline__ float sigmoid_f(float x) {
    // exact identity: sigmoid(x) = (1 + tanh(x/2)) / 2
    return 0.5f * fast_tanhf(0.5f * x) + 0.5f;
}

// ---------------------------------------------------------------------------
// Kernel A: hb[g][b][h] = sum_j h0[b][j] * Wh_g[h][j] + bi_g[h] + bh_g[h]
// 4096 threads (g,h); each keeps 32 batch accumulators so each Wh row is
// streamed exactly once (16 MB total read).
// ---------------------------------------------------------------------------
__global__ void __launch_bounds__(256)
lstm_hidden_bias_kernel(const float* __restrict__ h0,
                        const float* __restrict__ wh_i, const float* __restrict__ wh_f,
                        const float* __restrict__ wh_o, const float* __restrict__ wh_g,
                        const float* __restrict__ bii,  const float* __restrict__ bhi,
                        const float* __restrict__ bif,  const float* __restrict__ bhf,
                        const float* __restrict__ bio,  const float* __restrict__ bho,
                        const float* __restrict__ big,  const float* __restrict__ bhg,
                        float* __restrict__ hb)
{
    const int tid = blockIdx.x * blockDim.x + threadIdx.x;   // 0..4095
    const int g   = tid >> 10;                               // gate 0..3 (i,f,o,g)
    const int h   = tid & (H_SZ - 1);

    const float* W  = (g == 0) ? wh_i : (g == 1) ? wh_f : (g == 2) ? wh_o : wh_g;
    const float* bi = (g == 0) ? bii  : (g == 1) ? bif  : (g == 2) ? bio  : big;
    const float* bh = (g == 0) ? bhi  : (g == 1) ? bhf  : (g == 2) ? bho  : bhg;

    float acc[B_SZ];
#pragma unroll
    for (int b = 0; b < B_SZ; ++b) acc[b] = 0.0f;

    const float* Wrow = W + (size_t)h * K_SZ;
#pragma unroll 4
    for (int j = 0; j < K_SZ; ++j) {
        const float w = Wrow[j];
#pragma unroll
        for (int b = 0; b < B_SZ; ++b)
            acc[b] = fmaf(w, h0[b * K_SZ + j], acc[b]);
    }

    const float bias = bi[h] + bh[h];
#pragma unroll
    for (int b = 0; b < B_SZ; ++b)
        hb[((size_t)g * B_SZ + b) * H_SZ + h] = acc[b] + bias;
}

// ---------------------------------------------------------------------------
// Kernel B: fused 4-gate GEMM + LSTM epilogue.
// Per wave: 32(M)x16(N) tile, 4 gates -> 8 fp32 WMMA accumulators.
// A fragment (16x4 f32): lanes 0-15 hold (K0,K1), lanes 16-31 hold (K2,K3),
// i.e. per-lane float2 at  row*1024 + k + 2*(lane>=16).  B = W^T has the
// identical per-lane addressing on W's row-major storage.
// Per K=4 step: 2 A-loads + 4 B-loads feed 8 WMMAs.
// ---------------------------------------------------------------------------
__global__ void __launch_bounds__(256)
lstm_main_kernel(const float* __restrict__ X,
                 const float* __restrict__ wi, const float* __restrict__ wf,
                 const float* __restrict__ wo, const float* __restrict__ wg,
                 const float* __restrict__ c0,
                 const float* __restrict__ hb,
                 float* __restrict__ out)
{
    const int wave = threadIdx.x >> 5;
    const int lane = threadIdx.x & 31;
    const int lh   = lane >> 4;        // half-wave select (K pair / M+8 / N group)
    const int mn   = lane & 15;

    const int mt   = blockIdx.y;                 // M-tile (32 rows) 0..511
    const int nt   = blockIdx.x * 8 + wave;      // N-tile 0..63
    const int row0 = mt * 32;
    const int col0 = nt * 16;

    const int batch = mt >> 4;                   // 512 rows (=16 tiles) per batch
    const int tbase = (mt & 15) * 32;            // t of local row 0

    const size_t fragOff = (size_t)mn * K_SZ + 2 * lh;
    const float* Ap0 = X + (size_t)row0 * K_SZ + fragOff;          // rows m..m+15
    const float* Ap1 = Ap0 + (size_t)16 * K_SZ;                    // rows m+16..m+31
    const float* Bi  = wi + (size_t)col0 * K_SZ + fragOff;
    const float* Bf  = wf + (size_t)col0 * K_SZ + fragOff;
    const float* Bo  = wo + (size_t)col0 * K_SZ + fragOff;
    const float* Bg  = wg + (size_t)col0 * K_SZ + fragOff;

    v8f acc_i0 = {}; v8f acc_f0 = {}; v8f acc_o0 = {}; v8f acc_g0 = {};
    v8f acc_i1 = {}; v8f acc_f1 = {}; v8f acc_o1 = {}; v8f acc_g1 = {};

#pragma unroll 4
    for (int k = 0; k < K_SZ; k += 4) {
        const v2f a0 = *(const v2f*)(Ap0 + k);
        const v2f a1 = *(const v2f*)(Ap1 + k);
        const v2f bi = *(const v2f*)(Bi + k);
        const v2f bf = *(const v2f*)(Bf + k);
        const v2f bo = *(const v2f*)(Bo + k);
        const v2f bg = *(const v2f*)(Bg + k);
        acc_i0 = __builtin_amdgcn_wmma_f32_16x16x4_f32(false, a0, false, bi,
                                                       (short)0, acc_i0, false, false);
        acc_i1 = __builtin_amdgcn_wmma_f32_16x16x4_f32(false, a1, false, bi,
                                                       (short)0, acc_i1, false, false);
        acc_f0 = __builtin_amdgcn_wmma_f32_16x16x4_f32(false, a0, false, bf,
                                                       (short)0, acc_f0, false, false);
        acc_f1 = __builtin_amdgcn_wmma_f32_16x16x4_f32(false, a1, false, bf,
                                                       (short)0, acc_f1, false, false);
        acc_o0 = __builtin_amdgcn_wmma_f32_16x16x4_f32(false, a0, false, bo,
                                                       (short)0, acc_o0, false, false);
        acc_o1 = __builtin_amdgcn_wmma_f32_16x16x4_f32(false, a1, false, bo,
                                                       (short)0, acc_o1, false, false);
        acc_g0 = __builtin_amdgcn_wmma_f32_16x16x4_f32(false, a0, false, bg,
                                                       (short)0, acc_g0, false, false);
        acc_g1 = __builtin_amdgcn_wmma_f32_16x16x4_f32(false, a1, false, bg,
                                                       (short)0, acc_g1, false, false);
    }

    // ---- fused LSTM epilogue -------------------------------------------
    // C/D layout: VGPR v, lanes 0-15 -> M=v, N=lane; lanes 16-31 -> M=v+8.
    const int h = col0 + mn;                      // wave-private column
    const float hbi = hb[(size_t)(0 * B_SZ + batch) * H_SZ + h];
    const float hbf = hb[(size_t)(1 * B_SZ + batch) * H_SZ + h];
    const float hbo = hb[(size_t)(2 * B_SZ + batch) * H_SZ + h];
    const float hbg = hb[(size_t)(3 * B_SZ + batch) * H_SZ + h];
    const float c0v = c0[(size_t)batch * H_SZ + h];

    float* hid    = out;
    float* h_last = out + HID_ELEMS;
    float* c_last = out + HID_ELEMS + (size_t)B_SZ * H_SZ;

#pragma unroll
    for (int s = 0; s < 2; ++s) {                 // M half: rows +0 / +16
        const v8f ai = s ? acc_i1 : acc_i0;
        const v8f af = s ? acc_f1 : acc_f0;
        const v8f ao = s ? acc_o1 : acc_o0;
        const v8f ag = s ? acc_g1 : acc_g0;
#pragma unroll
        for (int v = 0; v < 8; ++v) {
            const int t = tbase + s * 16 + v + 8 * lh;
            const float iv = sigmoid_f(ai[v] + hbi);
            const float fv = sigmoid_f(af[v] + hbf);
            const float ov = sigmoid_f(ao[v] + hbo);
            const float gv = fast_tanhf(ag[v] + hbg);
            const float cv = fv * c0v + iv * gv;
            const float hv = ov * fast_tanhf(cv);
            hid[(size_t)t * (B_SZ * H_SZ) + (size_t)batch * H_SZ + h] = hv;
            if (t == T_SZ - 1) {
                h_last[(size_t)batch * H_SZ + h] = hv;
                c_last[(size_t)batch * H_SZ + h] = cv;
            }
        }
    }
}

// ---------------------------------------------------------------------------
extern "C" void kernel_launch(void* const* d_in, const int* in_sizes, int n_in,
                              void* d_out, int out_size, void* d_ws, size_t ws_size,
                              hipStream_t stream) {
    const float* inputs = (const float*)d_in[0];
    const float* h0     = (const float*)d_in[1];
    const float* c0     = (const float*)d_in[2];
    const float* w_ii = (const float*)d_in[3];  const float* b_ii = (const float*)d_in[4];
    const float* w_hi = (const float*)d_in[5];  const float* b_hi = (const float*)d_in[6];
    const float* w_if = (const float*)d_in[7];  const float* b_if = (const float*)d_in[8];
    const float* w_hf = (const float*)d_in[9];  const float* b_hf = (const float*)d_in[10];
    const float* w_io = (const float*)d_in[11]; const float* b_io = (const float*)d_in[12];
    const float* w_ho = (const float*)d_in[13]; const float* b_ho = (const float*)d_in[14];
    const float* w_ig = (const float*)d_in[15]; const float* b_ig = (const float*)d_in[16];
    const float* w_hg = (const float*)d_in[17]; const float* b_hg = (const float*)d_in[18];

    float* out = (float*)d_out;
    float* hb  = (float*)d_ws;   // [4][32][1024] f32 = 512 KB scratch

    // 1) hidden-state contribution + biases (tiny GEMM)
    lstm_hidden_bias_kernel<<<16, 256, 0, stream>>>(
        h0, w_hi, w_hf, w_ho, w_hg,
        b_ii, b_hi, b_if, b_hf, b_io, b_ho, b_ig, b_hg, hb);

    // 2) big fused 4-gate GEMM + activations + transpose-store
    dim3 grid(8, 512);   // x: 8 blocks x 8 waves = 64 N-tiles; y: 512 M-tiles
    lstm_main_kernel<<<grid, 256, 0, stream>>>(
        inputs, w_ii, w_if, w_io, w_ig, c0, hb, out);
}